// JaCDEManual_13829794693220
// MI455X (gfx1250) — compile-verified
//
#include <hip/hip_runtime.h>
#include <hip/hip_bf16.h>

// ---------------------------------------------------------------------------
// Neural-CDE RHS, restructured as a chain of (4096x128)x(128x128) fp32 GEMMs
// executed with V_WMMA_F32_16X16X4_F32 on gfx1250 (wave32).
//
//   u   = curr @ whT            (WMMA)
//   v   = drelu ⊙ u             (VALU, fragment-resident)
//   w   = v @ woutT             (WMMA)
//   curr= dtanh ⊙ w             (VALU)
// repeated 8x, plus the l1 / tanh / jx setup stages.
// ---------------------------------------------------------------------------

typedef __attribute__((ext_vector_type(2))) float v2f;
typedef __attribute__((ext_vector_type(8))) float v8f;

#define H 128
#define C 32
#define TILE_M 16     // batch rows per workgroup
#define LDW 132       // padded LDS row stride (floats) for 128-wide rows
#define XLDW 36       // padded LDS row stride for 32-wide rows

__device__ __forceinline__ v8f wmma4(v2f a, v2f b, v8f c) {
#if defined(__HIP_DEVICE_COMPILE__)
  // D = A(16x4,f32) * B(4x16,f32) + C(16x16,f32)
  // 8-arg form: (neg_a, A, neg_b, B, c_mod, C, reuse_a, reuse_b)
  return __builtin_amdgcn_wmma_f32_16x16x4_f32(
      false, a, false, b, (short)0, c, false, false);
#else
  return c;  // host pass: never executed
#endif
}

// A-matrix 16x4 fragment (ISA layout: lanes 0-15 hold M=0..15;
// VGPR0 = K 0 / K 2 for lane halves, VGPR1 = K 1 / K 3).
template <int SW>
__device__ __forceinline__ v2f load_afrag(const float A[][SW], int lane, int k0) {
  const int r  = lane & 15;
  const int kk = (lane >> 4) << 1;  // 0 for lanes 0-15, 2 for lanes 16-31
  v2f a;
  a.x = A[r][k0 + kk];
  a.y = A[r][k0 + kk + 1];
  return a;
}

// B-matrix 4x16 fragment from an LDS-resident transposed weight W[k][n].
__device__ __forceinline__ v2f load_bfrag(const float W[][LDW], int lane,
                                          int k0, int n0) {
  const int n  = n0 + (lane & 15);
  const int kk = (lane >> 4) << 1;
  v2f b;
  b.x = W[k0 + kk][n];
  b.y = W[k0 + kk + 1][n];
  return b;
}

// B-matrix fragment of wxT (K=32) read straight from global (L2-resident 16KB).
__device__ __forceinline__ v2f load_bfrag_wx(const float* __restrict__ wx,
                                             int lane, int k0, int n0) {
  const int n  = n0 + (lane & 15);
  const int kk = (lane >> 4) << 1;
  v2f b;
  b.x = wx[n * C + k0 + kk];
  b.y = wx[n * C + k0 + kk + 1];
  return b;
}

// Full K=128 GEMM stage for this wave's 16x16 tile: acc += A(16x128) * W(128x16)
template <int KTOT>
__device__ __forceinline__ v8f gemm_acc(const float A[][LDW],
                                        const float W[][LDW], int lane, int n0,
                                        v8f acc) {
#pragma unroll
  for (int k0 = 0; k0 < KTOT; k0 += 4) {
    acc = wmma4(load_afrag<LDW>(A, lane, k0), load_bfrag(W, lane, k0, n0), acc);
  }
  return acc;
}

__global__ __launch_bounds__(256) void cde_rhs_kernel(
    const float* __restrict__ t, const float* __restrict__ h,
    const float* __restrict__ coeffs, const float* __restrict__ dcoeffs,
    const float* __restrict__ tobs, const float* __restrict__ wx,
    const float* __restrict__ wh, const float* __restrict__ wout,
    const float* __restrict__ b0, const float* __restrict__ b1,
    float* __restrict__ out) {
  __shared__ float whT[H][LDW];    // whT[k][n]  = wh[n][k]
  __shared__ float woT[H][LDW];    // woT[k][n]  = wout[n][k]
  __shared__ float S0[TILE_M][LDW];  // activation ping
  __shared__ float S1[TILE_M][LDW];  // activation pong
  __shared__ float xb[TILE_M][XLDW];   // x(t)    tile
  __shared__ float xdb[TILE_M][XLDW];  // xdot(t) tile

  const int tid  = threadIdx.x;
  const int lane = tid & 31;
  const int wave = tid >> 5;
  const int n0   = wave * 16;              // wave-owned output columns
  const int rowBase = blockIdx.x * TILE_M; // batch rows of this block
  const int col  = n0 + (lane & 15);
  const int rOff = (lane >> 4) << 3;       // 0 or 8: D-layout row offset

  // --- spline interval: searchsorted(tobs, t, 'right') - 1, clipped ---------
  const float ts = t[0];
  int idx = 0;
#pragma unroll
  for (int i = 1; i < 17; ++i) idx += (tobs[i] <= ts) ? 1 : 0;
  idx = (idx > 15) ? 15 : idx;
  const float dt = ts - tobs[idx];

  // --- stage weights (transposed) + h tile + spline tiles into LDS ---------
  for (int i = tid; i < H * H; i += 256) {
    const int n = i >> 7, k = i & 127;
    whT[k][n] = wh[i];    // wh[n*128+k]
    woT[k][n] = wout[i];  // wout[n*128+k]
  }
  for (int i = tid; i < TILE_M * H; i += 256) {
    const int r = i >> 7, c = i & 127;
    S0[r][c] = h[(long)(rowBase + r) * H + c];
  }
  for (int i = tid; i < TILE_M * C; i += 256) {
    const int r = i >> 5, c = i & 31;
    const long base = ((long)(rowBase + r) * 16 + idx) * 128 + c; // (B,16,4,32)
    {
      const float c0 = coeffs[base],      c1 = coeffs[base + 32];
      const float c2 = coeffs[base + 64], c3 = coeffs[base + 96];
      xb[r][c] = c0 + dt * (c1 + dt * (c2 + dt * c3));
    }
    {
      const float c0 = dcoeffs[base],      c1 = dcoeffs[base + 32];
      const float c2 = dcoeffs[base + 64], c3 = dcoeffs[base + 96];
      xdb[r][c] = c0 + dt * (c1 + dt * (c2 + dt * c3));
    }
  }
  __syncthreads();

  v8f acc, drelu, dtanh, hdot;

  // --- l1 = x@wxT + h@whT + b0 ; relu -> S1 ; drelu kept in registers ------
  {
    const float bias = b0[col];
#pragma unroll
    for (int v = 0; v < 8; ++v) acc[v] = bias;
#pragma unroll
    for (int k0 = 0; k0 < C; k0 += 4)
      acc = wmma4(load_afrag<XLDW>(xb, lane, k0),
                  load_bfrag_wx(wx, lane, k0, n0), acc);
    acc = gemm_acc<H>(S0, whT, lane, n0, acc);
#pragma unroll
    for (int v = 0; v < 8; ++v) {
      const float l1 = acc[v];
      drelu[v] = 1.0f / (1.0f + __expf(-l1));      // sigmoid(l1)
      S1[v + rOff][col] = fmaxf(l1, 0.0f);         // relu(l1)
    }
  }
  __syncthreads();

  // --- tanh stage: dtanh = 1 - tanh(relu@woutT + b1)^2 ---------------------
  {
    const float bias = b1[col];
#pragma unroll
    for (int v = 0; v < 8; ++v) acc[v] = bias;
    acc = gemm_acc<H>(S1, woT, lane, n0, acc);
#pragma unroll
    for (int v = 0; v < 8; ++v) {
      const float th = tanhf(acc[v]);
      dtanh[v] = 1.0f - th * th;
    }
  }

  // --- jx = dtanh ⊙ ((drelu ⊙ (xdot@wxT)) @ woutT) -------------------------
  {
#pragma unroll
    for (int v = 0; v < 8; ++v) acc[v] = 0.0f;
#pragma unroll
    for (int k0 = 0; k0 < C; k0 += 4)
      acc = wmma4(load_afrag<XLDW>(xdb, lane, k0),
                  load_bfrag_wx(wx, lane, k0, n0), acc);
#pragma unroll
    for (int v = 0; v < 8; ++v) S0[v + rOff][col] = drelu[v] * acc[v];
    __syncthreads();
#pragma unroll
    for (int v = 0; v < 8; ++v) acc[v] = 0.0f;
    acc = gemm_acc<H>(S0, woT, lane, n0, acc);
#pragma unroll
    for (int v = 0; v < 8; ++v) {
      const float cur = dtanh[v] * acc[v];
      hdot[v] = cur;                 // h_dot = jx
      S1[v + rOff][col] = cur;       // curr -> S1
    }
    __syncthreads();
  }

  // --- 8-term Neumann series: curr = Jh·curr ; h_dot += curr ---------------
  for (int it = 0; it < 8; ++it) {
#pragma unroll
    for (int v = 0; v < 8; ++v) acc[v] = 0.0f;
    acc = gemm_acc<H>(S1, whT, lane, n0, acc);          // u = curr @ whT
#pragma unroll
    for (int v = 0; v < 8; ++v) S0[v + rOff][col] = drelu[v] * acc[v];
    __syncthreads();
#pragma unroll
    for (int v = 0; v < 8; ++v) acc[v] = 0.0f;
    acc = gemm_acc<H>(S0, woT, lane, n0, acc);          // w = v @ woutT
#pragma unroll
    for (int v = 0; v < 8; ++v) {
      const float cur = dtanh[v] * acc[v];
      hdot[v] += cur;
      S1[v + rOff][col] = cur;
    }
    __syncthreads();
  }

  // --- store h_dot tile ----------------------------------------------------
#pragma unroll
  for (int v = 0; v < 8; ++v)
    out[(long)(rowBase + v + rOff) * H + col] = hdot[v];
}

extern "C" void kernel_launch(void* const* d_in, const int* in_sizes, int n_in,
                              void* d_out, int out_size, void* d_ws,
                              size_t ws_size, hipStream_t stream) {
  const float* t       = (const float*)d_in[0];
  const float* h       = (const float*)d_in[1];
  const float* coeffs  = (const float*)d_in[2];
  const float* dcoeffs = (const float*)d_in[3];
  const float* tobs    = (const float*)d_in[4];
  const float* wx      = (const float*)d_in[5];
  const float* wh      = (const float*)d_in[6];
  const float* wout    = (const float*)d_in[7];
  const float* b0      = (const float*)d_in[8];
  const float* b1      = (const float*)d_in[9];
  float* out           = (float*)d_out;

  const int B = 4096;
  cde_rhs_kernel<<<B / TILE_M, 256, 0, stream>>>(
      t, h, coeffs, dcoeffs, tobs, wx, wh, wout, b0, b1, out);
}